// PropagationModule_26242250178804
// MI455X (gfx1250) — compile-verified
//
#include <hip/hip_runtime.h>

// Spatially-varying 9x9 aggregation (N=1, C=256, H=256, W=256)
//   out[c,h,w] = sum_{i,j} key[c, h+i-4, w+j-4] * wk[i*9+j, h, w]
// mapped onto V_WMMA_F32_16X16X4_F32 via a banded-matrix formulation.
// All zero-padding is folded into the banded B matrices (built once per
// block in LDS), so the compute loop is branch-free: A loads use clamped,
// always-valid addresses and rely on zeroed B rows to kill garbage terms.
// Per-lane 32-bit byte-offsets are kept opaque in VGPRs (empty asm
// constraint) so the backend selects SADDR+voffset global load/store forms
// instead of materializing 64-bit per-lane pointers.

#define CC   256
#define HH   256
#define WW   256
#define HWSZ (HH * WW)
#define KS   9
#define PAD  4
#define NKSTEP 6                 // K band = 24 = 6 steps of K=4
#define NFRAG  (KS * NKSTEP * 32)

typedef __attribute__((ext_vector_type(2))) float v2f;
typedef __attribute__((ext_vector_type(8))) float v8f;

__launch_bounds__(512)
__global__ void prop9x9_wmma_kernel(const float* __restrict__ key,
                                    const float* __restrict__ wgt,
                                    float* __restrict__ out)
{
    // Banded weight matrices in WMMA B-fragment layout:
    // frag (i, ks), lane L holds {B_i[ks*4 + 2*(L>>4), L&15], B_i[ks*4 + 2*(L>>4)+1, L&15]}
    __shared__ v2f lbs[NFRAG];

    const int w0   = blockIdx.x << 4;   // pixel-tile column base
    const int h    = blockIdx.y;        // output row
    const int tid  = threadIdx.x;
    const int lane = tid & 31;
    const int wave = tid >> 5;

    // ---- stage 1: cooperatively build the 9 banded B matrices ----
    // B_i[k, w'] = wk[i*9 + (k-w'), h, w0+w']  if 0<=k-w'<=8
    //              AND key row h+i-4 in range AND key col w0-4+k in range, else 0.
    for (int f = tid; f < NFRAG; f += 512) {
        const int  l    = f & 31;
        const int  fk   = f >> 5;
        const int  ks   = fk % NKSTEP;
        const int  i    = fk / NKSTEP;
        const int  row0 = (ks << 2) + ((l >> 4) << 1);
        const int  wc   = l & 15;
        const bool rv   = (unsigned)(h + i - PAD) < (unsigned)HH;
        const size_t pix = (size_t)h * WW + (size_t)(w0 + wc);
        v2f v;
#pragma unroll
        for (int t = 0; t < 2; ++t) {
            const int  k  = row0 + t;
            const int  j  = k - wc;
            const bool ok = rv && ((unsigned)j <= 8u) &&
                            ((unsigned)(w0 - PAD + k) < (unsigned)WW);
            const float val = ok ? wgt[(size_t)(i * KS + j) * HWSZ + pix] : 0.0f;
            if (t == 0) v.x = val; else v.y = val;
        }
        lbs[f] = v;
    }
    __syncthreads();

    // ---- stage 2: each wave owns a 16-channel x 16-pixel output tile ----
    const int c0    = wave << 4;
    const int m     = lane & 15;          // A-matrix row (channel within group)
    const int khalf = (lane >> 4) << 1;   // k-pair select within K=4 step

    // Per-lane 32-bit BYTE offsets (channel plane + clamped column),
    // invariant in i -> reused by all 9 rows as global-load voffsets.
    unsigned eoffb[NKSTEP];
#pragma unroll
    for (int ks = 0; ks < NKSTEP; ++ks) {
        int col = w0 - PAD + (ks << 2) + khalf;
        col = col < 0 ? 0 : (col > WW - 2 ? WW - 2 : col);   // stays 8B-aligned
        eoffb[ks] = (unsigned)(((c0 + m) * HWSZ + col) * 4);
    }

    v8f acc = {};
#pragma unroll
    for (int i = 0; i < KS; ++i) {
        int hh = h + i - PAD;                                // uniform
        hh = hh < 0 ? 0 : (hh > HH - 1 ? HH - 1 : hh);       // clamped row
        const char* rowbase = (const char*)(key + (size_t)hh * WW);  // SGPR base

        v2f a[NKSTEP], b[NKSTEP];
#pragma unroll
        for (int ks = 0; ks < NKSTEP; ++ks) {                // saddr+voffset b64 loads
            unsigned bo = eoffb[ks];
            asm("" : "+v"(bo));            // keep as 32-bit VGPR voffset
            a[ks] = *(const v2f*)(rowbase + bo);
        }
#pragma unroll
        for (int ks = 0; ks < NKSTEP; ++ks)                  // ds_load_2addr_b64 pairs
            b[ks] = lbs[(i * NKSTEP + ks) * 32 + lane];
#pragma unroll
        for (int ks = 0; ks < NKSTEP; ++ks)                  // 6x v_wmma per row
            acc = __builtin_amdgcn_wmma_f32_16x16x4_f32(
                      false, a[ks], false, b[ks], (short)0, acc, false, false);
    }

    // ---- stage 3: store C/D fragments ----
    // lane L holds D[mbase+v, n] in acc[v], n = L&15, mbase = (L>>4)*8.
    // One 32-bit voffset + immediate offsets v*HW*4 (<= 1.75MB, fits IOFFSET).
    const int n     = lane & 15;
    const int mbase = (lane >> 4) << 3;
    unsigned sob = (unsigned)((((c0 + mbase) * HWSZ) + h * WW + w0 + n) * 4);
    asm("" : "+v"(sob));                   // keep as 32-bit VGPR voffset
    char* ob = (char*)out;
#pragma unroll
    for (int v = 0; v < 8; ++v)
        *(float*)(ob + sob + (size_t)(v * (HWSZ * 4))) = acc[v];
}

extern "C" void kernel_launch(void* const* d_in, const int* in_sizes, int n_in,
                              void* d_out, int out_size, void* d_ws, size_t ws_size,
                              hipStream_t stream)
{
    (void)in_sizes; (void)n_in; (void)out_size; (void)d_ws; (void)ws_size;
    const float* key = (const float*)d_in[0];   // (1,256,256,256) f32
    const float* wgt = (const float*)d_in[1];   // (1,81,256,256)  f32
    float*       out = (float*)d_out;           // (1,256,256,256) f32

    dim3 grid(WW / 16, HH);   // one block per 16-wide pixel tile per row
    prop9x9_wmma_kernel<<<grid, 512, 0, stream>>>(key, wgt, out);
}